// WLHNEncoder_38096359915589
// MI455X (gfx1250) — compile-verified
//
#include <hip/hip_runtime.h>
#include <math.h>

typedef float v2f __attribute__((ext_vector_type(2)));
typedef float v8f __attribute__((ext_vector_type(8)));
typedef int v2i __attribute__((ext_vector_type(2)));
typedef __attribute__((address_space(1))) v2i as1_v2i;  // global
typedef __attribute__((address_space(3))) v2i as3_v2i;  // LDS

#define HDIM 128
#define MIN_NORM 1e-15f

#if __has_builtin(__builtin_amdgcn_global_load_async_to_lds_b64)
#define HAVE_ASYNC_LDS 1
#else
#define HAVE_ASYNC_LDS 0
#endif

// ---------------- helpers ----------------

__device__ __forceinline__ float wred32(float v) {
  v += __shfl_xor(v, 16, 32);
  v += __shfl_xor(v, 8, 32);
  v += __shfl_xor(v, 4, 32);
  v += __shfl_xor(v, 2, 32);
  v += __shfl_xor(v, 1, 32);
  return v;
}

__device__ __forceinline__ float dot4(const float4& a, const float4& b) {
  return a.x * b.x + a.y * b.y + a.z * b.z + a.w * b.w;
}
__device__ __forceinline__ float4 scale4(const float4& a, float s) {
  return make_float4(a.x * s, a.y * s, a.z * s, a.w * s);
}
__device__ __forceinline__ float4 sub4(const float4& a, const float4& b) {
  return make_float4(a.x - b.x, a.y - b.y, a.z - b.z, a.w - b.w);
}
__device__ __forceinline__ float4 fma4(const float4& a, float s, const float4& b) {
  return make_float4(fmaf(a.x, s, b.x), fmaf(a.y, s, b.y),
                     fmaf(a.z, s, b.z), fmaf(a.w, s, b.w));
}

// ---------------- WMMA GEMM: C[n,128] = act(A[n,128] @ W[128,128]^T + bias) ----------------
// 8 waves/block; W staged once per block into LDS (async DMA when available) in
// fragment-swizzled order: slot(nt,kk,lane) holds the v2f B-fragment that `lane`
// of a compute wave consumes for output tile nt at K-step kk. Reads are
// lane-contiguous ds_load_b64 -> bank-conflict-free.
// fp32 WMMA 16x16x4 fragment model:
//   A (v2f): {A[m][k0+off], A[m][k0+off+1]}, m=lane&15, off=(lane&16)?2:0
//   B (v2f): {W[col][k0+off], W[col][k0+off+1]}, col=nt*16+(lane&15)
//   C/D: vgpr r -> row r (lanes 0-15) / r+8 (lanes 16-31), col index = lane&15

__global__ void gemm128_wmma(const float* __restrict__ A, const float* __restrict__ W,
                             const float* __restrict__ bias, float* __restrict__ C,
                             int nTiles, int relu) {
  __shared__ __align__(16) float smem[16384];  // 64 KB: 8192 v2f fragment slots

  const int tid = threadIdx.x;
  const int wid = (blockIdx.x * blockDim.x + tid) >> 5;
  const int lane = tid & 31;
  const int m = lane & 15;
  const int koff = (lane & 16) ? 2 : 0;
  const int radd = (lane & 16) ? 8 : 0;
  const bool active = (wid < nTiles);                 // wave-uniform
  const size_t r0 = (size_t)(active ? wid : (nTiles - 1)) * 16;

  __builtin_prefetch(bias, 0, 0);

  // ---- stage W into LDS, fragment-swizzled (32 iters x 256 threads x 8B = 64KB)
  for (int i = 0; i < 32; ++i) {
    const int slot = i * 256 + tid;      // 0..8191
    const int sl_lane = slot & 31;
    const int sl_kk = (slot >> 5) & 31;
    const int sl_nt = slot >> 10;
    const int col = sl_nt * 16 + (sl_lane & 15);
    const int k = sl_kk * 4 + ((sl_lane & 16) ? 2 : 0);
    const float* g = W + (size_t)col * HDIM + k;
#if HAVE_ASYNC_LDS
    __builtin_amdgcn_global_load_async_to_lds_b64(
        (as1_v2i*)g, (as3_v2i*)(smem + slot * 2), 0, 0);
#else
    *(v2f*)(smem + slot * 2) = *(const v2f*)g;
#endif
  }
#if HAVE_ASYNC_LDS
#if __has_builtin(__builtin_amdgcn_s_wait_asynccnt)
  __builtin_amdgcn_s_wait_asynccnt(0);
#else
  asm volatile("s_wait_asynccnt 0" ::: "memory");
#endif
#endif
  __syncthreads();

  // ---- preload this wave's 16x128 A slab into registers (enables in-place C==A)
  const float* arow = A + (r0 + m) * HDIM + koff;
  v2f a[32];
#pragma unroll
  for (int kk = 0; kk < 32; ++kk) a[kk] = *(const v2f*)(arow + kk * 4);

  // ---- 8 output tiles x 32 K-steps of f32 WMMA, B fed from LDS
  for (int nt = 0; nt < 8; ++nt) {
    const float* bp = smem + nt * 2048 + lane * 2;
    v8f acc = {};
#pragma unroll
    for (int kk = 0; kk < 32; ++kk) {
      v2f b = *(const v2f*)(bp + kk * 64);
      acc = __builtin_amdgcn_wmma_f32_16x16x4_f32(false, a[kk], false, b,
                                                  (short)0, acc, false, false);
    }
    if (active) {                                     // wave-uniform store guard
      const int col = nt * 16 + m;
      const float bv = bias[col];
#pragma unroll
      for (int r = 0; r < 8; ++r) {
        float v = acc[r] + bv;
        if (relu) v = fmaxf(v, 0.0f);
        C[(r0 + (size_t)(r + radd)) * HDIM + col] = v;
      }
    }
  }
}

// ---------------- elementwise / rowwise kernels ----------------

__global__ void copy_f4(float* __restrict__ dst, const float* __restrict__ src, size_t n4) {
  size_t i = (size_t)blockIdx.x * blockDim.x + threadIdx.x;
  if (i < n4) ((float4*)dst)[i] = ((const float4*)src)[i];
}

// t[dst[e]] += h[src[e]]  (32 lanes per edge, 4 channels per lane)
__global__ void edge_scatter(const float* __restrict__ h, const int* __restrict__ src,
                             const int* __restrict__ dst, float* __restrict__ t, int e) {
  int idx = blockIdx.x * blockDim.x + threadIdx.x;
  int edge = idx >> 5;
  int lane = idx & 31;
  if (edge >= e) return;
  int s = src[edge], d = dst[edge];
  float4 v = *(const float4*)(h + (size_t)s * HDIM + lane * 4);
  float* o = t + (size_t)d * HDIM + lane * 4;
  atomicAdd(o + 0, v.x);
  atomicAdd(o + 1, v.y);
  atomicAdd(o + 2, v.z);
  atomicAdd(o + 3, v.w);
}

__global__ void bn_zero(float* stats) { stats[threadIdx.x] = 0.0f; }

// per-channel sum / sumsq over n rows -> stats[0:128], stats[128:256]
__global__ void bn_stats(const float* __restrict__ t, float* __restrict__ stats, int n) {
  const int c = threadIdx.x & 127;
  const int half = threadIdx.x >> 7;
  float s = 0.0f, s2 = 0.0f;
  for (int r = blockIdx.x * 2 + half; r < n; r += gridDim.x * 2) {
    float v = t[(size_t)r * HDIM + c];
    s += v;
    s2 += v * v;
  }
  __shared__ float sm[512];
  sm[threadIdx.x] = s;
  sm[256 + threadIdx.x] = s2;
  __syncthreads();
  if (half == 0) {
    s += sm[threadIdx.x + 128];
    s2 += sm[384 + threadIdx.x];
    atomicAdd(&stats[c], s);
    atomicAdd(&stats[128 + c], s2);
  }
}

__global__ void bn_apply(float* __restrict__ t, const float* __restrict__ stats,
                         const float* __restrict__ g, const float* __restrict__ bt,
                         int n, size_t total) {
  size_t i = (size_t)blockIdx.x * blockDim.x + threadIdx.x;
  if (i >= total) return;
  int c = (int)(i & 127);
  float invn = 1.0f / (float)n;
  float mean = stats[c] * invn;
  float var = stats[128 + c] * invn - mean * mean;
  t[i] = (t[i] - mean) * rsqrtf(var + 1e-5f) * g[c] + bt[c];
}

// z_prev = 0 ; z_cur = scaling * h / ||h||
__global__ void init_z(const float* __restrict__ h, float* __restrict__ zprev,
                       float* __restrict__ zcur, int n, float scaling) {
  int row = (blockIdx.x * blockDim.x + threadIdx.x) >> 5;
  int lane = threadIdx.x & 31;
  if (row >= n) return;
  size_t base = (size_t)row * HDIM + lane * 4;
  float4 hv = *(const float4*)(h + base);
  float hh = wred32(dot4(hv, hv));
  float inv = scaling / fmaxf(sqrtf(hh), 1e-30f);
  *(float4*)(zcur + base) = scale4(hv, inv);
  *(float4*)(zprev + base) = make_float4(0.f, 0.f, 0.f, 0.f);
}

// fused hyperbolic update: z_all from (h, z_prev, z_cur, p); writes zout (== z_prev buffer)
__global__ void z_update(const float* __restrict__ h, const float* __restrict__ zprev,
                         const float* __restrict__ zcur, const float* __restrict__ p,
                         float* __restrict__ zout, int n, float scaling) {
  int row = (blockIdx.x * blockDim.x + threadIdx.x) >> 5;
  int lane = threadIdx.x & 31;
  if (row >= n) return;
  size_t base = (size_t)row * HDIM + lane * 4;
  float4 hv = *(const float4*)(h + base);
  float4 zp = *(const float4*)(zprev + base);
  float4 zc = *(const float4*)(zcur + base);
  float4 pv = *(const float4*)(p + lane * 4);

  // z_ch = scaling * h / ||h||
  float hh = wred32(dot4(hv, hv));
  float4 zch = scale4(hv, scaling / fmaxf(sqrtf(hh), 1e-30f));

  // a = zc / max(<zc,zc>, MIN); r2 = <a,a> - 1
  float cc = wred32(dot4(zc, zc));
  float invc = 1.0f / fmaxf(cc, MIN_NORM);
  float4 a = scale4(zc, invc);
  float r2 = cc * invc * invc - 1.0f;

  // z_par = reflect_at_zero(z_prev, z_cur)
  float4 u = sub4(zp, a);
  float uu = wred32(dot4(u, u));
  float4 zpar = fma4(u, r2 / fmaxf(uu, MIN_NORM), a);

  // z_ch = reflect_through_zero(z_par, p, z_ch)
  float zp2 = wred32(dot4(zpar, zpar));
  float invzp = 1.0f / fmaxf(sqrtf(zp2), MIN_NORM);
  float pp = wred32(dot4(pv, pv));
  float invp = 1.0f / fmaxf(sqrtf(pp), MIN_NORM);
  float4 r = sub4(scale4(pv, invp), scale4(zpar, invzp));
  float rr = wred32(dot4(r, r));
  float rz = wred32(dot4(r, zch));
  float4 zch2 = fma4(r, -2.0f * (rz / rr), zch);

  // z_all = reflect_at_zero(z_ch, z_cur)
  float4 u2 = sub4(zch2, a);
  float uu2 = wred32(dot4(u2, u2));
  float4 zall = fma4(u2, r2 / fmaxf(uu2, MIN_NORM), a);

  *(float4*)(zout + base) = zall;
}

// out = y/||y|| * atanh(clip(||y||))
__global__ void logmap0_k(const float* __restrict__ y, float* __restrict__ o, int n) {
  int row = (blockIdx.x * blockDim.x + threadIdx.x) >> 5;
  int lane = threadIdx.x & 31;
  if (row >= n) return;
  size_t base = (size_t)row * HDIM + lane * 4;
  float4 v = *(const float4*)(y + base);
  float yy = wred32(dot4(v, v));
  float yn = fmaxf(sqrtf(yy), MIN_NORM);
  float t = fminf(yn, 0.99999994f);  // largest float < 1
  float s = atanhf(t) / yn;
  *(float4*)(o + base) = scale4(v, s);
}

__global__ void normalize_rows(float* __restrict__ o, int n) {
  int row = (blockIdx.x * blockDim.x + threadIdx.x) >> 5;
  int lane = threadIdx.x & 31;
  if (row >= n) return;
  size_t base = (size_t)row * HDIM + lane * 4;
  float4 v = *(float4*)(o + base);
  float nn = wred32(dot4(v, v));
  float inv = 1.0f / fmaxf(sqrtf(nn), 1e-12f);
  *(float4*)(o + base) = scale4(v, inv);
}

// ---------------- host launch ----------------

extern "C" void kernel_launch(void* const* d_in, const int* in_sizes, int n_in,
                              void* d_out, int out_size, void* d_ws, size_t ws_size,
                              hipStream_t stream) {
  const float* x = (const float*)d_in[0];
  const int* ei = (const int*)d_in[1];  // [2,E] int32
  const float* fc0_w = (const float*)d_in[2];
  const float* fc0_b = (const float*)d_in[3];
  const float* w1 = (const float*)d_in[4];
  const float* b1 = (const float*)d_in[5];
  const float* gamma = (const float*)d_in[6];
  const float* beta = (const float*)d_in[7];
  const float* w2 = (const float*)d_in[8];
  const float* b2 = (const float*)d_in[9];
  const float* fco_w = (const float*)d_in[10];
  const float* fco_b = (const float*)d_in[11];
  const float* p = (const float*)d_in[12];
  float* out = (float*)d_out;

  const int n = in_sizes[0] / HDIM;           // 40000 nodes
  const int e = in_sizes[1] / 2;              // 640000 edges
  const int L = in_sizes[4] / (HDIM * HDIM);  // 3 layers
  const size_t NH = (size_t)n * HDIM;

  float* h = (float*)d_ws;
  float* t = h + NH;
  float* zA = t + NH;
  float* zB = zA + NH;
  float* stats = zB + NH;  // 256 floats

  const float scaling = tanhf(0.5f);  // tanh(TAU/2)
  const int nTiles = n / 16;
  const dim3 blk(256);
  const int gemmGrid = (nTiles * 32 + 255) / 256;
  const int rowGrid = (n * 32 + 255) / 256;
  const int copyGrid = (int)((NH / 4 + 255) / 256);
  const int elemGrid = (int)((NH + 255) / 256);
  const int edgeGrid = (int)(((size_t)e * 32 + 255) / 256);

  const int* src = ei;
  const int* dst = ei + e;

  // h = relu(x @ fc0_w^T + fc0_b)
  gemm128_wmma<<<gemmGrid, blk, 0, stream>>>(x, fc0_w, fc0_b, h, nTiles, 1);
  // z_prev = 0; z_cur = scaling * h/||h||
  init_z<<<rowGrid, blk, 0, stream>>>(h, zA, zB, n, scaling);

  float* zPrev = zA;
  float* zCur = zB;
  for (int l = 0; l < L; ++l) {
    // t = h + segment_sum(h[src] -> dst)
    copy_f4<<<copyGrid, blk, 0, stream>>>(t, h, NH / 4);
    edge_scatter<<<edgeGrid, blk, 0, stream>>>(h, src, dst, t, e);
    // t = relu(t @ w1^T + b1)   (in-place safe: wave preloads its A slab)
    gemm128_wmma<<<gemmGrid, blk, 0, stream>>>(t, w1 + (size_t)l * HDIM * HDIM,
                                               b1 + (size_t)l * HDIM, t, nTiles, 1);
    // BatchNorm (training stats, biased var)
    bn_zero<<<1, 256, 0, stream>>>(stats);
    bn_stats<<<512, 256, 0, stream>>>(t, stats, n);
    bn_apply<<<elemGrid, blk, 0, stream>>>(t, stats, gamma + (size_t)l * HDIM,
                                           beta + (size_t)l * HDIM, n, NH);
    // h = relu(t @ w2^T + b2)
    gemm128_wmma<<<gemmGrid, blk, 0, stream>>>(t, w2 + (size_t)l * HDIM * HDIM,
                                               b2 + (size_t)l * HDIM, h, nTiles, 1);
    // hyperbolic tree-embedding update; z_all written into zPrev's buffer
    z_update<<<rowGrid, blk, 0, stream>>>(h, zPrev, zCur, p, zPrev, n, scaling);
    float* tmp = zPrev;
    zPrev = zCur;
    zCur = tmp;
  }

  // xh = logmap0(z_cur); out = xh @ fc_out^T + b; out /= ||out||
  logmap0_k<<<rowGrid, blk, 0, stream>>>(zCur, t, n);
  gemm128_wmma<<<gemmGrid, blk, 0, stream>>>(t, fco_w, fco_b, out, nTiles, 0);
  normalize_rows<<<rowGrid, blk, 0, stream>>>(out, n);
}